// LoRAMixin_62268435857672
// MI455X (gfx1250) — compile-verified
//
#include <hip/hip_runtime.h>

// LoRA forward for MI455X (gfx1250), fp32 end-to-end via V_WMMA_F32_16X16X4_F32.
// Global traffic fully coalesced through LDS; staging uses the CDNA5
// GLOBAL_LOAD_ASYNC_TO_LDS path (ASYNCcnt) with a 3-deep software pipeline,
// falling back to register-staged b128 copies otherwise.
//
//   inter[b,t,r] = sum_d x[b,t,d] * lora_A[idx[b], d, r]          (kernel 1 -> ws)
//   out[b,t,o]   = base[b,t,o] + s[idx[b]] * sum_r inter[b,t,r] * lora_B[idx[b], r, o]
//
// Shapes fixed by the reference: B=8, T=2048, D_IN=D_OUT=4096, R=64.

typedef __attribute__((ext_vector_type(2))) float v2f;
typedef __attribute__((ext_vector_type(4))) float v4f;
typedef __attribute__((ext_vector_type(8))) float v8f;
typedef int v4i_g __attribute__((vector_size(16)));   // matches builtin's pointee type

#define BATCH 8
#define T_DIM 2048
#define DIN   4096
#define DOUT  4096
#define RMAX  64

#define KC 32      // K-chunk staged per iteration in kernel 1
#define NBUF 3     // LDS pipeline depth (prefetch distance 2)
#define SX 36      // LDS row stride (floats) for x tile: conflict-free frag reads, 16B aligned
#define SA 72      // LDS row stride (floats) for lora_A/lora_B tiles: half-wave offset = 16 banks
#define SE 68      // LDS row stride (floats) for epilogue bounce: store conflict-free, b128 at floor

#if __has_builtin(__builtin_amdgcn_global_load_async_to_lds_b128) && \
    __has_builtin(__builtin_amdgcn_s_wait_asynccnt)
#define HAVE_ASYNC_LDS 1
#else
#define HAVE_ASYNC_LDS 0
#endif

#if HAVE_ASYNC_LDS
__device__ __forceinline__ void async_copy16(float* lds, const float* g) {
    // (global src, LDS dst, imm offset, cpol) — gfx1250 async global->LDS b128
    __builtin_amdgcn_global_load_async_to_lds_b128(
        (v4i_g*)g, (v4i_g*)lds, 0, 0);
}
#endif

// -------------------------------------------------------------------------
// Kernel 1: inter = x @ A
// Block: 256 threads = 8 waves, 4(M) x 2(N). Tile: 64 rows x 64 cols (all R).
// K loop: 4096 in chunks of 32; 3-buffer async pipeline, coalesced b128 staging.
// Grid: (T/64, B) = (32, 8).
// -------------------------------------------------------------------------
__global__ __launch_bounds__(256)
void lora_xA_kernel(const float* __restrict__ x,
                    const int*   __restrict__ adapter_indices,
                    const float* __restrict__ lora_A,
                    float*       __restrict__ inter)
{
    __shared__ float Xs[NBUF][64 * SX];   // 64 rows x KC, padded
    __shared__ float As[NBUF][KC * SA];   // KC rows x 64, padded

    const int b       = blockIdx.y;
    const int adapter = adapter_indices[b];

    const float* __restrict__ xb = x      + (size_t)b       * T_DIM * DIN;
    const float* __restrict__ Ab = lora_A + (size_t)adapter * DIN   * RMAX;
    float*       __restrict__ ib = inter  + (size_t)b       * T_DIM * RMAX;

    const int tid   = threadIdx.x;
    const int wave  = tid >> 5;
    const int lane  = tid & 31;
    const int laneM = lane & 15;
    const int half  = lane >> 4;

    const int rowBase = blockIdx.x * 64;
    const int waveRow = (wave >> 1) * 16;
    const int colBase = (wave & 1) * 32;

    v8f acc0 = {};
    v8f acc1 = {};

    // One 16x16x4 step chain over a staged chunk.
    auto compute = [&](int buf) {
        const float* __restrict__ xs = &Xs[buf][(waveRow + laneM) * SX + 2 * half];
        const float* __restrict__ as = &As[buf][(2 * half) * SA + colBase + laneM];
        #pragma unroll
        for (int ks = 0; ks < KC / 4; ++ks) {
            v2f a = *(const v2f*)(xs + ks * 4);       // rows striped per lane, K+2h
            const float* ap = as + ks * 4 * SA;
            v2f b0, b1;
            b0.x = ap[0];           // lora_A[k+2h,   col]
            b0.y = ap[SA];          // lora_A[k+2h+1, col]
            b1.x = ap[16];
            b1.y = ap[SA + 16];
            acc0 = __builtin_amdgcn_wmma_f32_16x16x4_f32(
                       false, a, false, b0, (short)0, acc0, false, false);
            acc1 = __builtin_amdgcn_wmma_f32_16x16x4_f32(
                       false, a, false, b1, (short)0, acc1, false, false);
        }
    };

    const int nChunks = DIN / KC;   // 128

    // Coalesced staging: x chunk 64x32 = 512 v4f (2/thread, 8 per 128B row),
    //                    A chunk 32x64 = 512 v4f (2/thread, 16 per 256B row).
#if HAVE_ASYNC_LDS
    auto stage_async = [&](int kc, int buf) {
        #pragma unroll
        for (int p = 0; p < 2; ++p) {
            const int i = p * 256 + tid;
            const int r = i >> 3, q = i & 7;
            async_copy16(&Xs[buf][r * SX + q * 4],
                         xb + (size_t)(rowBase + r) * DIN + kc + q * 4);
        }
        #pragma unroll
        for (int p = 0; p < 2; ++p) {
            const int i = p * 256 + tid;
            const int r = i >> 4, q = i & 15;
            async_copy16(&As[buf][r * SA + q * 4],
                         Ab + (size_t)(kc + r) * RMAX + q * 4);
        }
    };

    stage_async(0, 0);            // prefetch distance 2: chunks 0 and 1 in flight
    stage_async(KC, 1);

    int c = 0;
    // Steady state: 4 async ops per chunk per thread, in-order completion =>
    // allowing 4 outstanding means chunk c has fully landed.
    for (; c < nChunks - (NBUF - 1); ++c) {
        __builtin_amdgcn_s_wait_asynccnt(4);
        __syncthreads();                           // chunk c visible to all waves
        stage_async((c + 2) * KC, (c + 2) % NBUF); // overwrites buffer read at c-1
        compute(c % NBUF);
    }
    // Tail: nothing left to prefetch; drain fully.
    for (; c < nChunks; ++c) {
        __builtin_amdgcn_s_wait_asynccnt(0);
        __syncthreads();
        compute(c % NBUF);
    }
#else
    v4f xr[2], ar[2];
    auto gload = [&](int kc) {
        #pragma unroll
        for (int p = 0; p < 2; ++p) {
            const int i = p * 256 + tid;
            const int r = i >> 3, q = i & 7;
            xr[p] = *(const v4f*)(xb + (size_t)(rowBase + r) * DIN + kc + q * 4);
        }
        #pragma unroll
        for (int p = 0; p < 2; ++p) {
            const int i = p * 256 + tid;
            const int r = i >> 4, q = i & 15;
            ar[p] = *(const v4f*)(Ab + (size_t)(kc + r) * RMAX + q * 4);
        }
    };
    auto sstore = [&](int buf) {
        #pragma unroll
        for (int p = 0; p < 2; ++p) {
            const int i = p * 256 + tid;
            const int r = i >> 3, q = i & 7;
            *(v4f*)&Xs[buf][r * SX + q * 4] = xr[p];
        }
        #pragma unroll
        for (int p = 0; p < 2; ++p) {
            const int i = p * 256 + tid;
            const int r = i >> 4, q = i & 15;
            *(v4f*)&As[buf][r * SA + q * 4] = ar[p];
        }
    };
    gload(0);
    sstore(0);
    for (int c = 0; c < nChunks; ++c) {
        if (c + 1 < nChunks) gload((c + 1) * KC);   // overlap next chunk with compute
        __syncthreads();                            // buf[c%NBUF] visible to all waves
        compute(c % NBUF);
        if (c + 1 < nChunks) sstore((c + 1) % NBUF);
    }
#endif

    // D layout: VGPR j -> M = j (lanes 0-15) / j+8 (lanes 16-31), N = laneM.
    #pragma unroll
    for (int j = 0; j < 8; ++j) {
        const int row = rowBase + waveRow + j + half * 8;
        ib[(size_t)row * RMAX + colBase + laneM]      = acc0[j];
        ib[(size_t)row * RMAX + colBase + 16 + laneM] = acc1[j];
    }
}

// -------------------------------------------------------------------------
// Kernel 2: out = base + s * (inter @ B)
// Block: 256 threads = 8 waves stacked in M. Tile: 128 rows x 64 cols.
// lora_B 64x64 tile staged once in LDS; epilogue bounced through LDS so
// base/out stream as b128 (2 x 256B contiguous per instruction).
// Grid: (DOUT/64, T/128, B) = (64, 16, 8).
// -------------------------------------------------------------------------
__global__ __launch_bounds__(256)
void lora_By_kernel(const float* __restrict__ inter,
                    const float* __restrict__ base_output,
                    const int*   __restrict__ adapter_indices,
                    const float* __restrict__ lora_B,
                    const float* __restrict__ lora_scaling,
                    float*       __restrict__ out)
{
    __shared__ float Bs[RMAX * SA];        // 64 x 64 lora_B tile, padded
    __shared__ float Es[8 * 16 * SE];      // per-wave 16 x 64 epilogue bounce

    const int b       = blockIdx.z;
    const int adapter = adapter_indices[b];
    const float s     = lora_scaling[adapter];

    const float* __restrict__ ib = inter       + (size_t)b       * T_DIM * RMAX;
    const float* __restrict__ Bb = lora_B      + (size_t)adapter * RMAX  * DOUT;
    const float* __restrict__ bb = base_output + (size_t)b       * T_DIM * DOUT;
    float*       __restrict__ ob = out         + (size_t)b       * T_DIM * DOUT;

    const int tid   = threadIdx.x;
    const int wave  = tid >> 5;
    const int lane  = tid & 31;
    const int laneM = lane & 15;
    const int half  = lane >> 4;

    const int rowBase = blockIdx.y * 128 + wave * 16;
    const int colBase = blockIdx.x * 64;

    // Stage lora_B tile: 64 rows x 64 cols = 1024 v4f, 4 per thread,
    // 16 threads cover one 256B row -> fully coalesced.
    #pragma unroll
    for (int p = 0; p < 4; ++p) {
        const int i = p * 256 + tid;
        const int r = i >> 4, q = i & 15;
#if HAVE_ASYNC_LDS
        async_copy16(&Bs[r * SA + q * 4],
                     Bb + (size_t)r * DOUT + colBase + q * 4);
#else
        *(v4f*)&Bs[r * SA + q * 4] =
            *(const v4f*)(Bb + (size_t)r * DOUT + colBase + q * 4);
#endif
    }
#if HAVE_ASYNC_LDS
    __builtin_amdgcn_s_wait_asynccnt(0);
#endif
    __syncthreads();

    const float* __restrict__ arow = ib + (size_t)(rowBase + laneM) * RMAX + 2 * half;
    const float* __restrict__ bs   = &Bs[(2 * half) * SA + laneM];

    v8f acc[4] = {};

    #pragma unroll
    for (int k = 0; k < RMAX; k += 4) {
        v2f a = *(const v2f*)(arow + k);     // inter fragment (L2-resident)
        const float* bp = bs + k * SA;
        #pragma unroll
        for (int n = 0; n < 4; ++n) {
            v2f bf;
            bf.x = bp[n * 16];               // Bs[k+2h,   n*16+laneM]
            bf.y = bp[SA + n * 16];          // Bs[k+2h+1, n*16+laneM]
            acc[n] = __builtin_amdgcn_wmma_f32_16x16x4_f32(
                         false, a, false, bf, (short)0, acc[n], false, false);
        }
    }

    // Epilogue: bounce accumulators through wave-private LDS region, then
    // stream base/out as b128 (lanes cover 2 rows x 256B contiguous).
    float* __restrict__ es = &Es[wave * 16 * SE];
    #pragma unroll
    for (int n = 0; n < 4; ++n) {
        #pragma unroll
        for (int j = 0; j < 8; ++j) {
            es[(j + half * 8) * SE + n * 16 + laneM] = acc[n][j];
        }
    }
    // wave-private region: same-wave ds RAW ordered via DScnt by the compiler

    #pragma unroll
    for (int it = 0; it < 8; ++it) {
        const int lr = 2 * it + half;                 // local row 0..15
        const int cq = laneM * 4;                     // 4-float column group
        v4f av = *(const v4f*)&es[lr * SE + cq];
        const size_t off = (size_t)(rowBase + lr) * DOUT + colBase + cq;
        v4f bv = *(const v4f*)(bb + off);
        v4f ov = bv + s * av;
        *(v4f*)(ob + off) = ov;
    }
}

// -------------------------------------------------------------------------
// Harness entry point.
// d_in order: x, base_output, adapter_indices, lora_A, lora_B, lora_scaling
// d_ws: >= B*T*R*4 = 4 MB used for the intermediate activation.
// -------------------------------------------------------------------------
extern "C" void kernel_launch(void* const* d_in, const int* in_sizes, int n_in,
                              void* d_out, int out_size, void* d_ws, size_t ws_size,
                              hipStream_t stream) {
    const float* x               = (const float*)d_in[0];
    const float* base_output     = (const float*)d_in[1];
    const int*   adapter_indices = (const int*)  d_in[2];
    const float* lora_A          = (const float*)d_in[3];
    const float* lora_B          = (const float*)d_in[4];
    const float* lora_scaling    = (const float*)d_in[5];
    float*       out             = (float*)d_out;
    float*       inter           = (float*)d_ws;   // [B, T, RMAX] fp32

    dim3 g1(T_DIM / 64, BATCH);
    lora_xA_kernel<<<g1, 256, 0, stream>>>(x, adapter_indices, lora_A, inter);

    dim3 g2(DOUT / 64, T_DIM / 128, BATCH);
    lora_By_kernel<<<g2, 256, 0, stream>>>(inter, base_output, adapter_indices,
                                           lora_B, lora_scaling, out);
}